// spd_model_43705587204325
// MI455X (gfx1250) — compile-verified
//
#include <hip/hip_runtime.h>

// ---------------------------------------------------------------------------
// SPDNet autoencoder (bimap + ReEig x4) for MI455X / gfx1250.
// ReEig(Y) = U max(L,eps) U^T  ==  0.5*(M + |M|) + eps*I,  M = Y - eps*I,
// |M| = M * sign(M), sign via Newton-Schulz:  N <- 1.5 N - 0.5 N^3.
// Everything becomes batched f32 GEMM -> V_WMMA_F32_16X16X4_F32.
// GEMM: 64x64 macro-tile, K chunks of 16 double-buffered through LDS; tiles
// are fetched with GLOBAL_LOAD_ASYNC_TO_LDS_B128 (ASYNCcnt) when available,
// else coalesced b128 loads + ds stores. HBM-bound NS loop ~21 GB compulsory
// traffic @ 23.3 TB/s ~= 0.9 ms; WMMA f32 keeps eigen-rectification accurate.
// ---------------------------------------------------------------------------

typedef float v2f __attribute__((ext_vector_type(2)));
typedef float v8f __attribute__((ext_vector_type(8)));

#define BATCH     512
#define NS_ITERS  16
#define REEIG_EPS 1e-4f

#define BM 64
#define BN 64
#define BK 16
#define LDA_PAD 20   // A-tile LDS stride: 16B-aligned rows AND conflict-free
                     // (20*r mod 64 distinct for r=0..15 since 20*16 = 5*64)

#if __has_builtin(__builtin_amdgcn_global_load_async_to_lds_b128) && \
    __has_builtin(__builtin_amdgcn_s_wait_asynccnt)
#define HAS_ASYNC_LDS 1
#else
#define HAS_ASYNC_LDS 0
#endif

// Exact parameter types of the async-LDS builtin (from clang diagnostic):
//   param0: vector_size(16) int, addrspace(1) ("__device__")
//   param1: vector_size(16) int, addrspace(3) ("__shared__")
typedef int v4i_ __attribute__((vector_size(16)));
typedef __attribute__((address_space(1))) v4i_ async_gptr;
typedef __attribute__((address_space(3))) v4i_ async_lptr;

// Generic->LDS: low 32 bits of a generic shared pointer are the LDS offset.
__device__ __forceinline__ async_lptr* to_lds4(float* p)
{ return (async_lptr*)(unsigned int)(unsigned long long)p; }
__device__ __forceinline__ async_gptr* to_glb4(const float* p)
{ return (async_gptr*)(unsigned long long)p; }

// ---------------------------------------------------------------------------
// Batched WMMA GEMM:  C[b] = alpha * op(A[b]) * B[b] + beta * D[b]
//   A element (m,k) = A[b*sA + m*rsA + k*csA]  (rsA=1,csA=ld expresses A^T)
//   B element (k,n) = B[b*sB + k*ldb + n]
//   C/D are Mp x Np row-major, ld = Np.  sA/sB/sD = 0 => operand shared.
// Grid: (ceil(Np/64), ceil(Mp/64), BATCH); block: 256 (8 waves).
// Wave w -> 16x32 strip: rows (w>>1)*16, cols (w&1)*32 of the 64x64 tile.
// Kp/Mp/Np multiples of 16 (edge tiles zero-filled, stores guarded).
// WMMA executes unconditionally in-loop => EXEC all ones (ISA requirement).
// ---------------------------------------------------------------------------
__global__ __launch_bounds__(256)
void wmma_gemm_f32(const float* __restrict__ A, long sA, int rsA, int csA,
                   const float* __restrict__ Bm, long sB, int ldb,
                   const float* __restrict__ Dm, long sD, float beta,
                   float* __restrict__ Cm, long sC,
                   int Mp, int Np, int Kp, float alpha)
{
    __shared__ __align__(16) float lA[2][BM * LDA_PAD];  // 64 rows x 16 k, x2
    __shared__ __align__(16) float lB[2][BK * BN];       // 16 k x 64 n,  x2

    const int tid  = threadIdx.x;
    const int lane = tid & 31;
    const int wave = tid >> 5;           // 0..7
    const int wm   = wave >> 1;          // 0..3 : 16-row sub-tile
    const int wn   = wave & 1;           // 0..1 : 32-col sub-tile
    const int row  = lane & 15;
    const int hi   = lane >> 4;          // half-wave (K pair / M half)

    const int m0 = blockIdx.y * BM;
    const int n0 = blockIdx.x * BN;
    const int b  = blockIdx.z;

    const float* Ab = A  + (long)b * sA;
    const float* Bb = Bm + (long)b * sB;

    // LDS fill coordinates (256 threads move one float4 each per tile)
    const int arow = tid >> 2;           // 0..63
    const int acol = (tid & 3) * 4;      // 0,4,8,12
    const int brow = tid >> 4;           // 0..15
    const int bcol = (tid & 15) * 4;     // 0..60

    const bool vecA  = (csA == 1);
    const bool fullA = (m0 + BM <= Mp);  // block-uniform: async legal
    const bool fullB = (n0 + BN <= Np);

    // ---- tile fill helpers (async when whole tile is in range) ----
    auto fillA = [&](int buf, int k) {
        float* dst = &lA[buf][arow * LDA_PAD + acol];
#if HAS_ASYNC_LDS
        if (vecA && fullA) {
            const float* src = Ab + (long)(m0 + arow) * rsA + (k + acol);
            __builtin_amdgcn_global_load_async_to_lds_b128(to_glb4(src), to_lds4(dst), 0, 0);
            return;
        }
#endif
        float4 v = make_float4(0.f, 0.f, 0.f, 0.f);
        if (m0 + arow < Mp) {
            if (vecA) {
                v = *(const float4*)(Ab + (long)(m0 + arow) * rsA + (k + acol));
            } else {  // transposed weight (small GEMMs only)
                const float* p = Ab + (long)(m0 + arow) * rsA + (long)(k + acol) * csA;
                v.x = p[0]; v.y = p[csA]; v.z = p[2 * csA]; v.w = p[3 * csA];
            }
        }
        dst[0] = v.x; dst[1] = v.y; dst[2] = v.z; dst[3] = v.w;
    };
    auto fillB = [&](int buf, int k) {
        float* dst = &lB[buf][brow * BN + bcol];
#if HAS_ASYNC_LDS
        if (fullB) {
            const float* src = Bb + (long)(k + brow) * ldb + (n0 + bcol);
            __builtin_amdgcn_global_load_async_to_lds_b128(to_glb4(src), to_lds4(dst), 0, 0);
            return;
        }
#endif
        float4 v = make_float4(0.f, 0.f, 0.f, 0.f);
        if (n0 + bcol < Np) {
            v = *(const float4*)(Bb + (long)(k + brow) * ldb + (n0 + bcol));
        }
        *(float4*)dst = v;
    };

    v8f acc0 = {}, acc1 = {};

    const int nch = Kp / BK;
    fillA(0, 0);
    fillB(0, 0);
#if HAS_ASYNC_LDS
    __builtin_amdgcn_s_wait_asynccnt(0);
#endif
    __syncthreads();

    for (int i = 0; i < nch; ++i) {
        const int cur = i & 1;
        if (i + 1 < nch) {               // prefetch next chunk into other buffer
            fillA(cur ^ 1, (i + 1) * BK);
            fillB(cur ^ 1, (i + 1) * BK);
        }

        #pragma unroll
        for (int kk = 0; kk < BK; kk += 4) {
            const int ka = kk + 2 * hi;
            // A frag 16x4: lane -> A[wm*16+row][ka..ka+1] (contiguous b64)
            v2f af, bf0, bf1;
            const float* ap = &lA[cur][(wm * 16 + row) * LDA_PAD + ka];
            af.x = ap[0]; af.y = ap[1];
            // B frag 4x16: lane -> B[ka..ka+1][wn*32 + row (+16)]
            const float* bp = &lB[cur][ka * BN + wn * 32 + row];
            bf0.x = bp[0];  bf0.y = bp[BN];
            bf1.x = bp[16]; bf1.y = bp[BN + 16];
            acc0 = __builtin_amdgcn_wmma_f32_16x16x4_f32(false, af, false, bf0,
                                                         (short)0, acc0, false, false);
            acc1 = __builtin_amdgcn_wmma_f32_16x16x4_f32(false, af, false, bf1,
                                                         (short)0, acc1, false, false);
        }

#if HAS_ASYNC_LDS
        __builtin_amdgcn_s_wait_asynccnt(0);  // next-chunk async writes landed
#endif
        __syncthreads();                      // also fences compute vs. refill
    }

    // ---- guarded store (divergence only after all WMMAs) ----
    const float* Db = Dm ? (Dm + (long)b * sD) : nullptr;
    float*       Cb = Cm + (long)b * sC;
    #pragma unroll
    for (int r = 0; r < 8; ++r) {
        const int m = m0 + wm * 16 + hi * 8 + r;   // C layout: VGPR r -> M=hi*8+r
        if (m < Mp) {
            const int n = n0 + wn * 32 + row;
            if (n < Np) {
                float v = alpha * acc0[r];
                if (Db) v += beta * Db[(long)m * Np + n];
                Cb[(long)m * Np + n] = v;
            }
            const int n2 = n + 16;
            if (n2 < Np) {
                float v = alpha * acc1[r];
                if (Db) v += beta * Db[(long)m * Np + n2];
                Cb[(long)m * Np + n2] = v;
            }
        }
    }
}

// ---------------------------------------------------------------------------
// Elementwise helpers
// ---------------------------------------------------------------------------
__global__ void pad_weight_k(const float* __restrict__ src, int rows, int cols,
                             float* __restrict__ dst, int prows, int pcols)
{
    int idx = blockIdx.x * blockDim.x + threadIdx.x;
    if (idx >= prows * pcols) return;
    int r = idx / pcols, c = idx - r * pcols;
    dst[idx] = (r < rows && c < cols) ? src[r * cols + c] : 0.f;
}

__global__ void diag_shift_k(float* __restrict__ M, int dp, int d, float shift, long n)
{
    long idx = (long)blockIdx.x * blockDim.x + threadIdx.x;
    if (idx >= n) return;
    long b = idx / d;
    int  i = (int)(idx - b * d);
    M[b * (long)dp * dp + (long)i * dp + i] += shift;
}

// One block per matrix: scal[b] = 1/frobenius(M[b])
__global__ void fro_norm_k(const float* __restrict__ M, long per, float* __restrict__ scal)
{
    const int b = blockIdx.x;
    const float* p = M + (long)b * per;
    float s = 0.f;
    for (long i = threadIdx.x; i < per; i += blockDim.x) { float v = p[i]; s += v * v; }
    __shared__ float red[256];
    red[threadIdx.x] = s;
    __syncthreads();
    for (int off = 128; off > 0; off >>= 1) {
        if (threadIdx.x < off) red[threadIdx.x] += red[threadIdx.x + off];
        __syncthreads();
    }
    if (threadIdx.x == 0) scal[b] = rsqrtf(fmaxf(red[0], 1e-30f));
}

__global__ void scale_k(const float* __restrict__ M, const float* __restrict__ scal,
                        float* __restrict__ N, long per, long tot)
{
    long idx = (long)blockIdx.x * blockDim.x + threadIdx.x;
    if (idx >= tot) return;
    N[idx] = M[idx] * scal[idx / per];
}

// out = 0.5*(M + |M|) + eps*I  on the true d x d block (padding -> 0)
__global__ void combine_k(const float* __restrict__ M, const float* __restrict__ Abs,
                          int dp, int d, float* __restrict__ dstPad,
                          float* __restrict__ dstOut, float eps, long tot)
{
    long idx = (long)blockIdx.x * blockDim.x + threadIdx.x;
    if (idx >= tot) return;
    const long per = (long)dp * dp;
    const long b = idx / per;
    const long r = idx - b * per;
    const int  m = (int)(r / dp);
    const int  n = (int)(r - (long)m * dp);
    const bool in = (m < d) && (n < d);
    float v = 0.f;
    if (in) {
        v = 0.5f * (M[idx] + Abs[idx]);
        if (m == n) v += eps;
    }
    if (dstPad) dstPad[idx] = v;
    if (dstOut && in) dstOut[(long)b * d * d + (long)m * d + n] = v;
}

// ---------------------------------------------------------------------------
// Host-side sequencing (all launches on `stream`; graph-capture safe)
// ---------------------------------------------------------------------------
static inline void gemm(hipStream_t st, const float* A, long sA, int rsA, int csA,
                        const float* B, long sB, int ldb,
                        const float* D, long sD, float beta,
                        float* C, long sC, int Mp, int Np, int Kp, float alpha)
{
    dim3 g((Np + BN - 1) / BN, (Mp + BM - 1) / BM, BATCH), blk(256);
    wmma_gemm_f32<<<g, blk, 0, st>>>(A, sA, rsA, csA, B, sB, ldb, D, sD, beta,
                                     C, sC, Mp, Np, Kp, alpha);
}

// ReEig via Newton-Schulz sign iteration. M in place gets Y - eps*I.
// R0/R1/R2: rotating scratch (avoids in-place GEMM read/write races).
static void reeig_seq(hipStream_t st, float* M, float* R0, float* R1, float* R2,
                      int dp, int d, float* scal, float* dstPad, float* dstOut)
{
    const long per = (long)dp * dp;
    const long tot = per * BATCH;

    {   // M = Y - eps*I
        long n = (long)BATCH * d;
        diag_shift_k<<<(int)((n + 255) / 256), 256, 0, st>>>(M, dp, d, -REEIG_EPS, n);
    }
    fro_norm_k<<<BATCH, 256, 0, st>>>(M, per, scal);
    scale_k<<<(int)((tot + 255) / 256), 256, 0, st>>>(M, scal, R0, per, tot);

    float* P = R0;  // current N
    float* Q = R1;  // N^2 scratch
    float* R = R2;  // next N
    for (int i = 0; i < NS_ITERS; ++i) {
        // Q = P * P
        gemm(st, P, per, dp, 1, P, per, dp, nullptr, 0, 0.f, Q, per, dp, dp, dp, 1.0f);
        // R = 1.5*P - 0.5*P*Q
        gemm(st, P, per, dp, 1, Q, per, dp, P, per, 1.5f, R, per, dp, dp, dp, -0.5f);
        float* t = P; P = R; R = Q; Q = t;   // rotate: new N in P
    }
    // |M| = M * sign(M)  -> Q (free buffer)
    gemm(st, M, per, dp, 1, P, per, dp, nullptr, 0, 0.f, Q, per, dp, dp, dp, 1.0f);
    combine_k<<<(int)((tot + 255) / 256), 256, 0, st>>>(M, Q, dp, d, dstPad, dstOut,
                                                        REEIG_EPS, tot);
}

extern "C" void kernel_launch(void* const* d_in, const int* in_sizes, int n_in,
                              void* d_out, int out_size, void* d_ws, size_t ws_size,
                              hipStream_t stream)
{
    (void)in_sizes; (void)n_in; (void)out_size; (void)ws_size;

    const float* x      = (const float*)d_in[0];  // [512,400,400]
    const float* w_enc  = (const float*)d_in[1];  // [400,100]
    const float* w_enc1 = (const float*)d_in[2];  // [100,50]
    const float* w_dec1 = (const float*)d_in[3];  // [50,100]
    const float* w_dec  = (const float*)d_in[4];  // [100,400]
    float* out = (float*)d_out;                   // X1 | X2 | X4 concat
    float* ws  = (float*)d_ws;

    // ---- workspace arena (float offsets). Padded dims: 100->112, 50->64. ----
    float* const W1p  = ws + 0;          // [400,112]
    float* const W2p  = ws + 44800;      // [112, 64]
    float* const W3p  = ws + 51968;      // [ 64,112]
    float* const W4p  = ws + 59136;      // [112,400]
    float* const scal = ws + 103936;     // [512]
    float* const X1p  = ws + 104448;     // [512,112,112]
    float* const X2p  = ws + 6526976;    // [512, 64, 64]
    float* const X3p  = ws + 8624128;    // [512,112,112]
    float* const TT   = ws + 15046656;   // [512,400,112] intermediate
    float* const BG0  = ws + 37984256;   // [512,400,400] M
    float* const BG1  = ws + 119904256;  // [512,400,400] NS rot
    float* const BG2  = ws + 201824256;  // [512,400,400] NS rot
    float* const BG3  = ws + 283744256;  // [512,400,400] NS rot

    // ---- zero-pad weights to 16-multiples (padding is algebraically inert) --
    pad_weight_k<<<(400 * 112 + 255) / 256, 256, 0, stream>>>(w_enc,  400, 100, W1p, 400, 112);
    pad_weight_k<<<(112 *  64 + 255) / 256, 256, 0, stream>>>(w_enc1, 100,  50, W2p, 112,  64);
    pad_weight_k<<<( 64 * 112 + 255) / 256, 256, 0, stream>>>(w_dec1,  50, 100, W3p,  64, 112);
    pad_weight_k<<<(112 * 400 + 255) / 256, 256, 0, stream>>>(w_dec,  100, 400, W4p, 112, 400);

    // ---- Stage 1: Y1 = W1^T x W1 ; X1 = ReEig(Y1) ----
    gemm(stream, x,   160000L, 400, 1,   W1p, 0, 112, nullptr, 0, 0.f, TT,  44800L, 400, 112, 400, 1.f);
    gemm(stream, W1p, 0, 1, 112,         TT, 44800L, 112, nullptr, 0, 0.f, BG0, 12544L, 112, 112, 400, 1.f);
    reeig_seq(stream, BG0, BG1, BG2, BG3, 112, 100, scal, X1p, out);               // out[0..5.12M)

    // ---- Stage 2: Y2 = W2^T X1 W2 ; X2 = ReEig(Y2) ----
    gemm(stream, X1p, 12544L, 112, 1,    W2p, 0, 64, nullptr, 0, 0.f, TT,  7168L, 112,  64, 112, 1.f);
    gemm(stream, W2p, 0, 1, 64,          TT,  7168L, 64, nullptr, 0, 0.f, BG0, 4096L,  64,  64, 112, 1.f);
    reeig_seq(stream, BG0, BG1, BG2, BG3, 64, 50, scal, X2p, out + 5120000);       // X2 block

    // ---- Stage 3: Y3 = W3^T X2 W3 ; X3 = ReEig(Y3) (internal only) ----
    gemm(stream, X2p, 4096L, 64, 1,      W3p, 0, 112, nullptr, 0, 0.f, TT,  7168L,  64, 112,  64, 1.f);
    gemm(stream, W3p, 0, 1, 112,         TT,  7168L, 112, nullptr, 0, 0.f, BG0, 12544L, 112, 112,  64, 1.f);
    reeig_seq(stream, BG0, BG1, BG2, BG3, 112, 100, scal, X3p, nullptr);

    // ---- Stage 4: Y4 = W4^T X3 W4 ; X4 = ReEig(Y4) ----
    gemm(stream, X3p, 12544L, 112, 1,    W4p, 0, 400, nullptr, 0, 0.f, TT,  44800L, 112, 400, 112, 1.f);
    gemm(stream, W4p, 0, 1, 400,         TT,  44800L, 400, nullptr, 0, 0.f, BG0, 160000L, 400, 400, 112, 1.f);
    reeig_seq(stream, BG0, BG1, BG2, BG3, 400, 400, scal, nullptr, out + 6400000); // X4 block
}